// LanePooling_32323923870244
// MI455X (gfx1250) — compile-verified
//
#include <hip/hip_runtime.h>
#include <math.h>

typedef __attribute__((ext_vector_type(2))) float v2f;
typedef __attribute__((ext_vector_type(8))) float v8f;

#define WPAIR 144  // float2 pitch per k-pair in LDS weight layout (2*144=288 floats; 288%64=32 -> halves hit disjoint banks)

__device__ __forceinline__ v8f wmma4(v2f a, v2f b, v8f c) {
  // D = A(16x4 f32) * B(4x16 f32) + C(16x16 f32)
  return __builtin_amdgcn_wmma_f32_16x16x4_f32(false, a, false, b, (short)0, c, false, false);
}

// Stage [K,128] row-major weight into LDS as pair-interleaved:
// WL[p][n] = (W[2p][n], W[2p+1][n]) at float2 index p*WPAIR + n
__device__ __forceinline__ void stage_w(float* dst, const float* __restrict__ W, int K) {
  const int tid = threadIdx.x, nt = blockDim.x;
  const int total = (K >> 1) * 128;
  for (int idx = tid; idx < total; idx += nt) {
    const int p = idx >> 7, n = idx & 127;
    v2f v;
    v.x = W[(p * 2) * 128 + n];
    v.y = W[(p * 2 + 1) * 128 + n];
    *(v2f*)(dst + (p * WPAIR + n) * 2) = v;
  }
}

// GEMM: A[16,K] from LDS staging (row pitch `pitch` floats) times staged weights -> C-layout regs
__device__ __forceinline__ void gemm_tile(v8f acc[8], const float* st, int pitch,
                                          const float* wl, int K, int half, int c) {
#pragma unroll
  for (int j = 0; j < 8; ++j) {
    v8f z = {0.f, 0.f, 0.f, 0.f, 0.f, 0.f, 0.f, 0.f};
    acc[j] = z;
  }
  const float* arow  = st + c * pitch + 2 * half;
  const float* wbase = wl + half * (2 * WPAIR) + 2 * c;
  for (int k0 = 0; k0 < K; k0 += 4) {
    v2f a = *(const v2f*)(arow + k0);
    const float* wp = wbase + (k0 >> 1) * (2 * WPAIR);
#pragma unroll
    for (int j = 0; j < 8; ++j) {
      v2f b = *(const v2f*)(wp + j * 32);
      acc[j] = wmma4(a, b, acc[j]);
    }
  }
}

// GroupNorm(1 group over 128 ch) + optional ReLU on a C-layout tile.
// Row m = r + 8*half lives entirely in one 16-lane half -> xor-reduce masks 1,2,4,8.
__device__ __forceinline__ void gn_tile(v8f acc[8], const float g[8], const float b[8], bool relu) {
#pragma unroll
  for (int r = 0; r < 8; ++r) {
    float s = 0.f, q = 0.f;
#pragma unroll
    for (int j = 0; j < 8; ++j) { float v = acc[j][r]; s += v; q = fmaf(v, v, q); }
#pragma unroll
    for (int m = 1; m <= 8; m <<= 1) { s += __shfl_xor(s, m); q += __shfl_xor(q, m); }
    const float mu = s * (1.f / 128.f);
    const float rs = rsqrtf(q * (1.f / 128.f) - mu * mu + 1e-5f);
#pragma unroll
    for (int j = 0; j < 8; ++j) {
      float v = (acc[j][r] - mu) * rs * g[j] + b[j];
      acc[j][r] = relu ? fmaxf(v, 0.f) : v;
    }
  }
}

// Dump C-layout regs into LDS A-staging: st[m*pitch + colbase + n]
__device__ __forceinline__ void dump_tile(const v8f acc[8], float* st, int pitch,
                                          int colbase, int half, int c) {
#pragma unroll
  for (int r = 0; r < 8; ++r) {
    float* row = st + (r + 8 * half) * pitch + colbase + c;
#pragma unroll
    for (int j = 0; j < 8; ++j) row[j * 16] = acc[j][r];
  }
}

__device__ __forceinline__ void load8(float d[8], const float* __restrict__ src, int c) {
#pragma unroll
  for (int j = 0; j < 8; ++j) d[j] = src[j * 16 + c];
}

// ------------------- Kernel 1: out = target_feat @ W_in -------------------
__global__ void __launch_bounds__(256) k_init(const float* __restrict__ tgt,
                                              const float* __restrict__ W_in,
                                              float* __restrict__ out, int Nt) {
  __shared__ float wl[64 * 2 * WPAIR];     // 73.7 KB
  __shared__ float stg[8][16 * 132];       // 67.6 KB
  stage_w(wl, W_in, 128);
  __syncthreads();
  const int lane = threadIdx.x & 31, wv = threadIdx.x >> 5;
  const int half = lane >> 4, c = lane & 15;
  float* st = stg[wv];
  const int ntiles = (Nt + 15) >> 4;
  const int wave = blockIdx.x * 8 + wv, nwaves = gridDim.x * 8;
  for (int tile = wave; tile < ntiles; tile += nwaves) {
    const int base = tile << 4;
    for (int m = 0; m < 16; ++m) {
      int row = base + m; if (row >= Nt) row = Nt - 1;
      *(float4*)(st + m * 132 + lane * 4) = *(const float4*)(tgt + (size_t)row * 128 + lane * 4);
    }
    v8f acc[8];
    gemm_tile(acc, st, 132, wl, 128, half, c);
#pragma unroll
    for (int r = 0; r < 8; ++r) {
      const int row = base + r + 8 * half;
      if (row < Nt) {
        float* o = out + (size_t)row * 128 + c;
#pragma unroll
        for (int j = 0; j < 8; ++j) o[j * 16] = acc[j][r];
      }
    }
  }
}

// --------- Kernel 2: per-edge ctx branch, scatter-add into out ---------
__global__ void __launch_bounds__(128) k_edge(
    const float* __restrict__ ctx_feat, const float* __restrict__ ctx_pose,
    const float* __restrict__ tgt_pose, const int* __restrict__ hi,
    const int* __restrict__ wi, const float* __restrict__ W_rp,
    const float* __restrict__ b_rp, const float* __restrict__ W1,
    const float* __restrict__ g_ctx, const float* __restrict__ be_ctx,
    const float* __restrict__ W2, float* __restrict__ out, int E) {
  __shared__ float wl1[128 * 2 * WPAIR];   // K=256, 147.5 KB
  __shared__ float wl2[64 * 2 * WPAIR];    // K=128, 73.7 KB
  __shared__ float stg[4][16 * 260];       // 66.6 KB
  stage_w(wl1, W1, 256);
  stage_w(wl2, W2, 128);
  __syncthreads();
  const int lane = threadIdx.x & 31, wv = threadIdx.x >> 5;
  const int half = lane >> 4, c = lane & 15;
  float* st = stg[wv];
  v2f brp[8]; float bias[8], gg[8], gb[8];
#pragma unroll
  for (int j = 0; j < 8; ++j) {
    brp[j].x = W_rp[(2 * half) * 128 + j * 16 + c];
    brp[j].y = W_rp[(2 * half + 1) * 128 + j * 16 + c];
    bias[j] = b_rp[j * 16 + c];
    gg[j] = g_ctx[j * 16 + c];
    gb[j] = be_ctx[j * 16 + c];
  }
  const int ntiles = (E + 15) >> 4;
  const int wave = blockIdx.x * 4 + wv, nwaves = gridDim.x * 4;
  for (int tile = wave; tile < ntiles; tile += nwaves) {
    const int base = tile << 4;
    // dist_feat = relu([16,4] posedelta @ W_rp + b_rp): single WMMA per n-tile (K=4)
    {
      int e = base + c; const bool val = e < E; if (!val) e = E - 1;
      const float* cp = ctx_pose + (size_t)hi[e] * 4;
      const float* tp = tgt_pose + (size_t)wi[e] * 4;
      v2f a;
      a.x = val ? (cp[2 * half] - tp[2 * half]) : 0.f;
      a.y = val ? (cp[2 * half + 1] - tp[2 * half + 1]) : 0.f;
      v8f dacc[8];
#pragma unroll
      for (int j = 0; j < 8; ++j) {
        v8f cm;
#pragma unroll
        for (int r = 0; r < 8; ++r) cm[r] = bias[j];
        dacc[j] = wmma4(a, brp[j], cm);
      }
#pragma unroll
      for (int r = 0; r < 8; ++r) {
        float* row = st + (r + 8 * half) * 260 + 128 + c;
#pragma unroll
        for (int j = 0; j < 8; ++j) row[j * 16] = fmaxf(dacc[j][r], 0.f);
      }
    }
    // gather context_feat rows -> staging cols 0..127 (coalesced float4)
    for (int m = 0; m < 16; ++m) {
      const int e = base + m;
      if (e < E) {
        const float* cf = ctx_feat + (size_t)hi[e] * 128;
        *(float4*)(st + m * 260 + lane * 4) = *(const float4*)(cf + lane * 4);
      } else {
        float4 z = {0.f, 0.f, 0.f, 0.f};
        *(float4*)(st + m * 260 + lane * 4) = z;
      }
    }
    v8f acc[8];
    gemm_tile(acc, st, 260, wl1, 256, half, c);   // [16,256] @ W_ctx1
    gn_tile(acc, gg, gb, true);
    dump_tile(acc, st, 260, 0, half, c);
    gemm_tile(acc, st, 260, wl2, 128, half, c);   // [16,128] @ W_ctx2
#pragma unroll
    for (int r = 0; r < 8; ++r) {
      const int e = base + r + 8 * half;
      if (e < E) {
        float* o = out + (size_t)wi[e] * 128 + c;
#pragma unroll
        for (int j = 0; j < 8; ++j) atomicAdd(o + j * 16, acc[j][r]);
      }
    }
  }
}

// --------- Kernel 3: node MLP tail, in-place on out ---------
__global__ void __launch_bounds__(256) k_node(
    const float* __restrict__ tgt, const float* __restrict__ g_n, const float* __restrict__ be_n,
    const float* __restrict__ W1, const float* __restrict__ g1, const float* __restrict__ b1,
    const float* __restrict__ W2, const float* __restrict__ g2, const float* __restrict__ b2,
    float* __restrict__ out, int Nt) {
  __shared__ float wl1[64 * 2 * WPAIR];
  __shared__ float wl2[64 * 2 * WPAIR];
  __shared__ float stg[8][16 * 132];
  stage_w(wl1, W1, 128);
  stage_w(wl2, W2, 128);
  __syncthreads();
  const int lane = threadIdx.x & 31, wv = threadIdx.x >> 5;
  const int half = lane >> 4, c = lane & 15;
  float* st = stg[wv];
  float gnv[8], gnb[8], g1v[8], b1v[8], g2v[8], b2v[8];
  load8(gnv, g_n, c); load8(gnb, be_n, c);
  load8(g1v, g1, c);  load8(b1v, b1, c);
  load8(g2v, g2, c);  load8(b2v, b2, c);
  const int ntiles = (Nt + 15) >> 4;
  const int wave = blockIdx.x * 8 + wv, nwaves = gridDim.x * 8;
  for (int tile = wave; tile < ntiles; tile += nwaves) {
    const int base = tile << 4;
    v8f acc[8];
#pragma unroll
    for (int r = 0; r < 8; ++r) {
      int row = base + r + 8 * half; if (row >= Nt) row = Nt - 1;
      const float* p = out + (size_t)row * 128 + c;
#pragma unroll
      for (int j = 0; j < 8; ++j) acc[j][r] = p[j * 16];
    }
    gn_tile(acc, gnv, gnb, true);
    dump_tile(acc, st, 132, 0, half, c);
    gemm_tile(acc, st, 132, wl1, 128, half, c);   // @ W_m1
    gn_tile(acc, g1v, b1v, true);
    dump_tile(acc, st, 132, 0, half, c);
    gemm_tile(acc, st, 132, wl2, 128, half, c);   // @ W_m2
    gn_tile(acc, g2v, b2v, false);
#pragma unroll
    for (int r = 0; r < 8; ++r) {
      const int row = base + r + 8 * half;
      if (row < Nt) {
        const float* idp = tgt + (size_t)row * 128 + c;
        float* o = out + (size_t)row * 128 + c;
#pragma unroll
        for (int j = 0; j < 8; ++j) o[j * 16] = fmaxf(acc[j][r] + idp[j * 16], 0.f);
      }
    }
  }
}

extern "C" void kernel_launch(void* const* d_in, const int* in_sizes, int n_in,
                              void* d_out, int out_size, void* d_ws, size_t ws_size,
                              hipStream_t stream) {
  (void)n_in; (void)out_size; (void)d_ws; (void)ws_size;
  const float* ctx_feat = (const float*)d_in[0];
  const float* tgt_feat = (const float*)d_in[1];
  const float* ctx_pose = (const float*)d_in[2];
  const float* tgt_pose = (const float*)d_in[3];
  const int*   hi       = (const int*)d_in[4];
  const int*   wi       = (const int*)d_in[5];
  const float* W_in     = (const float*)d_in[6];
  const float* W_rp     = (const float*)d_in[7];
  const float* b_rp     = (const float*)d_in[8];
  const float* W_c1     = (const float*)d_in[9];
  const float* g_ctx    = (const float*)d_in[10];
  const float* be_ctx   = (const float*)d_in[11];
  const float* W_c2     = (const float*)d_in[12];
  const float* g_n      = (const float*)d_in[13];
  const float* be_n     = (const float*)d_in[14];
  const float* W_m1     = (const float*)d_in[15];
  const float* g_m1     = (const float*)d_in[16];
  const float* be_m1    = (const float*)d_in[17];
  const float* W_m2     = (const float*)d_in[18];
  const float* g_m2     = (const float*)d_in[19];
  const float* be_m2    = (const float*)d_in[20];
  float* out = (float*)d_out;

  const int Nt = in_sizes[1] / 128;
  const int E  = in_sizes[4];

  const int t1 = (Nt + 15) / 16;
  int grid1 = (t1 + 7) / 8; if (grid1 < 1) grid1 = 1;
  k_init<<<grid1, 256, 0, stream>>>(tgt_feat, W_in, out, Nt);

  const int t2 = (E + 15) / 16;
  int grid2 = (t2 + 3) / 4; if (grid2 > 2048) grid2 = 2048; if (grid2 < 1) grid2 = 1;
  k_edge<<<grid2, 128, 0, stream>>>(ctx_feat, ctx_pose, tgt_pose, hi, wi,
                                    W_rp, b_rp, W_c1, g_ctx, be_ctx, W_c2, out, E);

  k_node<<<grid1, 256, 0, stream>>>(tgt_feat, g_n, be_n, W_m1, g_m1, be_m1,
                                    W_m2, g_m2, be_m2, out, Nt);
}